// SoftSkeletonize_3289944949074
// MI455X (gfx1250) — compile-verified
//
#include <hip/hip_runtime.h>

#ifndef __has_builtin
#define __has_builtin(x) 0
#endif

// Problem constants (match reference: B=16, C=1, H=W=1024, 40 iterations)
#define NBATCH   16
#define IMG_H    1024
#define IMG_W    1024
#define NUM_ITER 40

// Tiling: 64x64 output tile, halo 3 (erode -> erode -> dilate chain).
// S0 stages a 70-row x 72-col window (cols gx = bx-4 .. bx+67) so interior
// tiles can be filled with 16B-aligned b128 async DMAs.
#define TS    64
#define D0    70               // rows staged (gy = by-3 .. by+66)
#define W0    72               // cols staged (gx = bx-4 .. bx+67), 16B aligned
#define STR0  72               // LDS row stride for S0 (16B-aligned rows)
#define D1    68               // after first erode (logical 68x68)
#define D2    66               // after second erode (logical 66x66)
#define STR1  (D1 + 1)
#define STR2  (D2 + 1)
#define NT    256              // 8 wave32s per block
#define NQ    (D0 * (W0 / 4))  // 1260 b128 quads per tile

// Address-space typed pointers for the gfx1250 async global->LDS builtins.
// Probe: b32 takes '__device__ int *'; b128 takes a pointer to
// 'int __attribute__((vector_size(16)))' in the same address spaces.
typedef int v4i_t __attribute__((__vector_size__(16)));
typedef __attribute__((address_space(1))) int   gint_t;
typedef __attribute__((address_space(3))) int   lint_t;
typedef __attribute__((address_space(1))) v4i_t gv4i_t;
typedef __attribute__((address_space(3))) v4i_t lv4i_t;

__device__ __forceinline__ void async_ld_b128(const float* gp, float* lp) {
#if defined(__HIP_DEVICE_COMPILE__) && \
    __has_builtin(__builtin_amdgcn_global_load_async_to_lds_b128)
  __builtin_amdgcn_global_load_async_to_lds_b128(
      (gv4i_t*)const_cast<float*>(gp), (lv4i_t*)lp, /*offset=*/0, /*cpol=*/0);
#elif defined(__HIP_DEVICE_COMPILE__) && \
    __has_builtin(__builtin_amdgcn_global_load_async_to_lds_b32)
  __builtin_amdgcn_global_load_async_to_lds_b32(
      (gint_t*)const_cast<float*>(gp + 0), (lint_t*)(lp + 0), 0, 0);
  __builtin_amdgcn_global_load_async_to_lds_b32(
      (gint_t*)const_cast<float*>(gp + 1), (lint_t*)(lp + 1), 0, 0);
  __builtin_amdgcn_global_load_async_to_lds_b32(
      (gint_t*)const_cast<float*>(gp + 2), (lint_t*)(lp + 2), 0, 0);
  __builtin_amdgcn_global_load_async_to_lds_b32(
      (gint_t*)const_cast<float*>(gp + 3), (lint_t*)(lp + 3), 0, 0);
#else
  *(float4*)lp = *(const float4*)gp;  // fallback
#endif
}

__device__ __forceinline__ void wait_async() {
#if defined(__HIP_DEVICE_COMPILE__) && \
    (__has_builtin(__builtin_amdgcn_global_load_async_to_lds_b128) || \
     __has_builtin(__builtin_amdgcn_global_load_async_to_lds_b32))
#if __has_builtin(__builtin_amdgcn_s_wait_asynccnt)
  __builtin_amdgcn_s_wait_asynccnt(0);
#else
  asm volatile("s_wait_asynccnt 0" ::: "memory");
#endif
#endif
}

// One fused soft-skeleton step.
// mode == 0 (init):  S1 = im;        sk = relu(im - dilate(erode(im)))
// mode == 1 (step):  S1 = erode(im); sk += relu(d - sk*d),
//                    d = relu(S1 - dilate(erode(S1))), im_dst = S1
__global__ __launch_bounds__(NT) void soft_skel_kernel(
    const float* __restrict__ im_src, float* __restrict__ im_dst,
    float* sk, int mode) {
  __shared__ __align__(16) float S0[D0 * STR0];
  __shared__ float S1[D1 * STR1];
  __shared__ float S2[D2 * STR2];

  const int tid = threadIdx.x;
  const int bx = blockIdx.x * TS;
  const int by = blockIdx.y * TS;
  const size_t plane = (size_t)blockIdx.z * (size_t)(IMG_H * IMG_W);
  const float* img = im_src + plane;
  const float PINF = __builtin_inff();

  const bool interior = (blockIdx.x > 0) && (blockIdx.x < gridDim.x - 1) &&
                        (blockIdx.y > 0) && (blockIdx.y < gridDim.y - 1);

  // ---- Stage 0: stage im tile into LDS -------------------------------------
  // Logical column j0 in [0,70) (gx = bx-3+j0) lives at physical col j0+1.
  if (interior) {
    // Whole halo in-bounds: 16B-aligned async global->LDS b128 DMAs.
    for (int idx = tid; idx < NQ; idx += NT) {
      int i = idx / (W0 / 4), q = idx - i * (W0 / 4);
      const float* gp =
          img + (size_t)(by - 3 + i) * IMG_W + (bx - 4) + 4 * q;
      async_ld_b128(gp, &S0[i * STR0 + 4 * q]);
    }
    wait_async();
  } else {
    // Edge tiles: guarded loads; out-of-image = +inf (min-pool identity).
    for (int idx = tid; idx < D0 * D0; idx += NT) {
      int i = idx / D0, j = idx - i * D0;
      int gy = by - 3 + i, gx = bx - 3 + j;
      float v = PINF;
      if (gy >= 0 && gy < IMG_H && gx >= 0 && gx < IMG_W)
        v = img[(size_t)gy * IMG_W + gx];
      S0[i * STR0 + j + 1] = v;
    }
  }
  __syncthreads();

  // ---- Stage 1: S1 = erode(S0) (cross min), or copy in init mode -----------
  for (int idx = tid; idx < D1 * D1; idx += NT) {
    int i = idx / D1, j = idx - i * D1;
    int c = (i + 1) * STR0 + (j + 2);  // physical center for logical (i+1,j+1)
    float v;
    if (mode) {
      v = fminf(fminf(fminf(S0[c - STR0], S0[c + STR0]),
                      fminf(S0[c - 1], S0[c + 1])),
                S0[c]);
      int gy = by - 2 + i, gx = bx - 2 + j;
      if (gy < 0 || gy >= IMG_H || gx < 0 || gx >= IMG_W) v = PINF;
    } else {
      v = S0[c];  // out-of-image already +inf
    }
    S1[i * STR1 + j] = v;
  }
  __syncthreads();

  // ---- Stage 2: S2 = erode(S1); out-of-image -> -inf (dilate identity) -----
  for (int idx = tid; idx < D2 * D2; idx += NT) {
    int i = idx / D2, j = idx - i * D2;
    int c = (i + 1) * STR1 + (j + 1);
    float v = fminf(fminf(fminf(S1[c - STR1], S1[c + STR1]),
                          fminf(S1[c - 1], S1[c + 1])),
                    S1[c]);
    int gy = by - 1 + i, gx = bx - 1 + j;
    if (gy < 0 || gy >= IMG_H || gx < 0 || gx >= IMG_W) v = -PINF;
    S2[i * STR2 + j] = v;
  }
  __syncthreads();

  // ---- Stage 3: open = dilate3x3(S2); delta; skel update; im write ---------
  // Each thread handles 4 consecutive pixels -> float4 (b128) global traffic.
  for (int k = 0; k < 4; ++k) {
    int i = (tid >> 4) + 16 * k;
    int j = (tid & 15) * 4;
    int c0 = i * STR2 + j;
    int c1 = c0 + STR2;
    int c2 = c1 + STR2;
    float open[4], imc[4];
#pragma unroll
    for (int m = 0; m < 4; ++m) {
      float h0 = fmaxf(fmaxf(S2[c0 + m], S2[c0 + m + 1]), S2[c0 + m + 2]);
      float h1 = fmaxf(fmaxf(S2[c1 + m], S2[c1 + m + 1]), S2[c1 + m + 2]);
      float h2 = fmaxf(fmaxf(S2[c2 + m], S2[c2 + m + 1]), S2[c2 + m + 2]);
      open[m] = fmaxf(fmaxf(h0, h1), h2);
      imc[m] = S1[(i + 2) * STR1 + (j + 2 + m)];
    }
    size_t g = plane + (size_t)(by + i) * IMG_W + (bx + j);
    if (mode) {
      float4 s4 = *(const float4*)(sk + g);
      float4 o4, w4;
      float* sp = (float*)&s4;
      float* op = (float*)&o4;
      float* wp = (float*)&w4;
#pragma unroll
      for (int m = 0; m < 4; ++m) {
        float delta = fmaxf(imc[m] - open[m], 0.0f);
        op[m] = sp[m] + fmaxf(delta - sp[m] * delta, 0.0f);
        wp[m] = imc[m];
      }
      *(float4*)(im_dst + g) = w4;
      *(float4*)(sk + g) = o4;
    } else {
      float4 o4;
      float* op = (float*)&o4;
#pragma unroll
      for (int m = 0; m < 4; ++m)
        op[m] = fmaxf(imc[m] - open[m], 0.0f);
      *(float4*)(sk + g) = o4;
    }
  }
}

extern "C" void kernel_launch(void* const* d_in, const int* in_sizes, int n_in,
                              void* d_out, int out_size, void* d_ws,
                              size_t ws_size, hipStream_t stream) {
  const float* img = (const float*)d_in[0];
  float* sk = (float*)d_out;
  const size_t nelem = (size_t)NBATCH * IMG_H * IMG_W;
  float* wsa = (float*)d_ws;          // im ping
  float* wsb = wsa + nelem;           // im pong (needs ws >= 128 MB)

  dim3 grid(IMG_W / TS, IMG_H / TS, NBATCH);
  dim3 block(NT);

  // skel0 = relu(img - open(img)); im untouched (d_in stays const)
  soft_skel_kernel<<<grid, block, 0, stream>>>(img, nullptr, sk, 0);

  const float* src = img;
  float* dst = wsa;
  for (int k = 0; k < NUM_ITER; ++k) {
    soft_skel_kernel<<<grid, block, 0, stream>>>(src, dst, sk, 1);
    src = dst;
    dst = (dst == wsa) ? wsb : wsa;
  }
}